// StreamingRhythmProjector_25254407700700
// MI455X (gfx1250) — compile-verified
//
#include <hip/hip_runtime.h>
#include <stdint.h>

// Problem geometry (fixed by the reference).
#define U_DIM 4096
#define T 256                         // threads per block (8 wave32s)
#define NW (T / 32)                   // waves per block
#define CPT (U_DIM / (T * 4))         // float4 chunks per thread per row = 4
#define EPT (U_DIM / T)               // scalar elems per thread per row = 16

// Dynamic-LDS layout (byte offsets; no static __shared__, so dynamic base = 0)
#define LDS_SCORES 0                  // 16 KB: w row, overwritten by scores
#define LDS_B      16384              // 16 KB: boundary row
#define LDS_PREV   32768              // 16 KB: previous_pause_exec row
#define LDS_HIST   49152              // 1 KB : 256-bin radix histogram
#define LDS_RED    50176              // NW floats: reduction scratch
#define LDS_STATE  (LDS_RED + NW * 4) // 2 uints: radix {k_rem, prefix}
#define LDS_TOTAL  (LDS_STATE + 16)

typedef float __attribute__((ext_vector_type(4))) f32x4;

__device__ __forceinline__ float wave_sum(float v) {
#pragma unroll
  for (int o = 16; o > 0; o >>= 1) v += __shfl_xor(v, o, 32);
  return v;
}

__device__ __forceinline__ float block_sum(float v, float* s_red, int tid) {
  v = wave_sum(v);
  __syncthreads();                       // protect s_red reuse across calls
  if ((tid & 31) == 0) s_red[tid >> 5] = v;
  __syncthreads();
  if (tid == 0) {
    float t = 0.f;
#pragma unroll
    for (int i = 0; i < NW; ++i) t += s_red[i];
    s_red[0] = t;
  }
  __syncthreads();
  return s_red[0];
}

// gfx1250 async global->LDS copy, 16B/lane, non-temporal (stream-once data).
__device__ __forceinline__ void async_b128_nt(uint32_t lds_off, uint64_t gaddr) {
  asm volatile("global_load_async_to_lds_b128 %0, %1, off th:TH_LOAD_NT"
               :: "v"(lds_off), "v"(gaddr) : "memory");
}

__global__ __launch_bounds__(T) void streaming_rhythm_projector_kernel(
    const float* __restrict__ w, const float* __restrict__ bsc,
    const float* __restrict__ mask, const float* __restrict__ budget,
    const float* __restrict__ prev, const int* __restrict__ frontier,
    float* __restrict__ out) {
  extern __shared__ char dyn_lds[];
  float*    s_scores = (float*)(dyn_lds + LDS_SCORES);
  float*    s_b      = (float*)(dyn_lds + LDS_B);
  float*    s_prev   = (float*)(dyn_lds + LDS_PREV);
  unsigned* s_hist   = (unsigned*)(dyn_lds + LDS_HIST);
  float*    s_red    = (float*)(dyn_lds + LDS_RED);
  unsigned* s_state  = (unsigned*)(dyn_lds + LDS_STATE);

  const int    row  = blockIdx.x;
  const int    tid  = threadIdx.x;
  const size_t base = (size_t)row * U_DIM;

  // ---- stage 1: async global->LDS copies. Issue order matters: w,b first,
  // prev last, so s_wait_asynccnt 4 (in-order completion) covers w+b while
  // prev is still in flight during score computation + radix select. --------
  const uint64_t wa = (uint64_t)(w + base);
  const uint64_t ba = (uint64_t)(bsc + base);
  const uint64_t pa = (uint64_t)(prev + base);
#pragma unroll
  for (int i = 0; i < CPT; ++i) {
    const uint32_t off16 = (uint32_t)(tid + i * T) * 16u;
    async_b128_nt(LDS_SCORES + off16, wa + off16);
    async_b128_nt(LDS_B + off16, ba + off16);
  }
#pragma unroll
  for (int i = 0; i < CPT; ++i) {
    const uint32_t off16 = (uint32_t)(tid + i * T) * 16u;
    async_b128_nt(LDS_PREV + off16, pa + off16);
  }

  // ---- overlap: stream mask (used once) and reduce -> visible ------------
  const f32x4* m4 = (const f32x4*)(mask + base);
  float visp = 0.f;
#pragma unroll
  for (int i = 0; i < CPT; ++i) {
    f32x4 m = __builtin_nontemporal_load(&m4[tid + i * T]);
    visp += m.x + m.y + m.z + m.w;
  }
  const float vis_f   = block_sum(visp, s_red, tid);  // exact integer count
  const int   visible = (int)vis_f;

  // keep_k = clamp(round(visible*0.35), 1, max(visible,1))
  float keep_f = fmaxf(1.f, rintf(vis_f * 0.35f));    // RNE, matches jnp.round
  keep_f = fminf(keep_f, fmaxf(vis_f, 1.f));
  const bool no_gate = (keep_f >= vis_f);

  // w + b complete once ASYNCcnt <= 4 (the 4 prev ops issued last).
  asm volatile("s_wait_asynccnt 0x4" ::: "memory");
  __syncthreads();

  // ---- stage 2: scores = (max(w,0) + 0.15*(0.1+max(b,0))) * mask, in LDS -
#pragma unroll
  for (int i = 0; i < EPT; ++i) {
    const int col = tid + i * T;
    float sc = 0.f;
    if (col < visible) {
      sc = fmaxf(s_scores[col], 0.f) + 0.15f * (0.1f + fmaxf(s_b[col], 0.f));
    }
    s_scores[col] = sc;                  // overwrite w buffer with scores
  }
  __syncthreads();

  // ---- stage 3: exact k-th largest via 4x8-bit radix select over LDS -----
  // All visible scores >= 0.015 > 0 => IEEE bit pattern order == value order.
  unsigned prefix = 0u;
  unsigned k_rem  = (unsigned)(int)keep_f;
  if (!no_gate) {                        // row-uniform branch: barriers safe
    for (int pass = 0; pass < 4; ++pass) {
      const int shift = 24 - 8 * pass;
      s_hist[tid] = 0u;
      __syncthreads();
#pragma unroll
      for (int i = 0; i < EPT; ++i) {
        const int col = tid + i * T;
        if (col < visible) {
          const unsigned u = __float_as_uint(s_scores[col]);
          const bool match =
              (pass == 0) || ((u >> (shift + 8)) == (prefix >> (shift + 8)));
          if (match) atomicAdd(&s_hist[(u >> shift) & 0xFFu], 1u);
        }
      }
      __syncthreads();
      if (tid == 0) {
        unsigned c = 0u, sel = 0u, kk = k_rem;
        for (int d = 255; d >= 0; --d) {
          const unsigned h = s_hist[d];
          if (c + h >= kk) { sel = (unsigned)d; kk -= c; break; }
          c += h;
        }
        s_state[0] = kk;
        s_state[1] = prefix | (sel << shift);
      }
      __syncthreads();
      k_rem  = s_state[0];
      prefix = s_state[1];
    }
  }
  const float thr = __uint_as_float(prefix);

  // prev row must be in LDS before stage 4 reads s_prev.
  asm volatile("s_wait_asynccnt 0x0" ::: "memory");
  __syncthreads();

  // ---- stage 4: prefix/tail budget allocation ----------------------------
  int vf = frontier[row];
  vf = min(max(vf, 0), U_DIM);
  const int pf_end = min(vf, visible);

  float ps = 0.f;                        // sum(prev) over frozen prefix
#pragma unroll
  for (int i = 0; i < EPT; ++i) {
    const int col = tid + i * T;
    if (col < pf_end) ps += s_prev[col];
  }
  const float prefix_sum = block_sum(ps, s_red, tid);

  const int   tail_n    = max(visible - vf, 0);
  const float tail_sum  = (float)tail_n;                 // exact (ints < 2^24)
  const float feps      = 1e-6f / fmaxf(tail_sum, 1.f);  // fallback*EPS on tail
  const float remaining = fmaxf(budget[row] - prefix_sum, 0.f);
  const float inv_temp  = 1.0f / 0.12f;

  float cs = 0.f;                        // sum of candidates over tail
#pragma unroll
  for (int i = 0; i < EPT; ++i) {
    const int col = tid + i * T;
    if (col >= vf && col < visible) {
      const float sc = s_scores[col];
      float sp = sc;
      if (!no_gate) sp = sc / (1.f + __expf(-(sc - thr) * inv_temp));
      cs += fmaxf(sp, 0.f) + feps;
    }
  }
  const float total = fmaxf(block_sum(cs, s_red, tid), 1e-6f);
  const float scale = remaining / total;

  // ---- stage 5: write output (nontemporal, stream-once) ------------------
  f32x4* o4 = (f32x4*)(out + base);
#pragma unroll
  for (int i = 0; i < CPT; ++i) {
    const int c4 = tid + i * T;
    f32x4 o;
#pragma unroll
    for (int j = 0; j < 4; ++j) {
      const int col = c4 * 4 + j;
      float v = 0.f;
      if (col < pf_end) {
        v = s_prev[col];                 // frozen prefix (mask==1 here)
      } else if (col >= vf && col < visible) {
        const float sc = s_scores[col];
        float sp = sc;
        if (!no_gate) sp = sc / (1.f + __expf(-(sc - thr) * inv_temp));
        v = (fmaxf(sp, 0.f) + feps) * scale;   // tail_n>0 implied here
      }
      if (j == 0) o.x = v; else if (j == 1) o.y = v;
      else if (j == 2) o.z = v; else o.w = v;
    }
    __builtin_nontemporal_store(o, &o4[c4]);
  }
}

extern "C" void kernel_launch(void* const* d_in, const int* in_sizes, int n_in,
                              void* d_out, int out_size, void* d_ws, size_t ws_size,
                              hipStream_t stream) {
  const float* w        = (const float*)d_in[0];
  const float* bsc      = (const float*)d_in[1];
  const float* mask     = (const float*)d_in[2];
  const float* budget   = (const float*)d_in[3];
  const float* prev     = (const float*)d_in[4];
  const int*   frontier = (const int*)d_in[5];
  float*       out      = (float*)d_out;

  const int B = in_sizes[3];            // rows (budget is [B])
  dim3 grid(B), block(T);
  streaming_rhythm_projector_kernel<<<grid, block, LDS_TOTAL, stream>>>(
      w, bsc, mask, budget, prev, frontier, out);
}